// GCN_KDE_14766097563857
// MI455X (gfx1250) — compile-verified
//
#include <hip/hip_runtime.h>
#include <math.h>

// ---------------------------------------------------------------------------
// GCN + GlobalAttention + soft-quantile KDE pipeline for MI455X (gfx1250).
// fp32 throughout; matmuls use V_WMMA_F32_16X16X4_F32 (wave32 WMMA).
// KDE inner loops restructured: 3 VALU + 1 TRANS per exp, quantiles
// register-blocked (single sweep over the grid, 40 accumulator VGPRs).
// ---------------------------------------------------------------------------

typedef __attribute__((ext_vector_type(2))) float v2f;
typedef __attribute__((ext_vector_type(8))) float v8f;

#if __has_builtin(__builtin_amdgcn_wmma_f32_16x16x4_f32)
#define HAS_WMMA_F32X4 1
#else
#define HAS_WMMA_F32X4 0
#endif

namespace {
constexpr int NN   = 2048;   // nodes
constexpr int D    = 64;     // feature dim (D_in == D_h)
constexpr int GG   = 16;     // graphs
constexpr int NPG  = 128;    // nodes per graph
constexpr int PP   = 500;    // kde grid points
constexpr int QQ   = 20;     // quantiles
constexpr int DOUT = 16;     // output dim
constexpr float SQRT2PI  = 2.5066282746310002f;
constexpr float NPOW     = 0.37892914162759955f;   // 128^(-0.2), Silverman factor
constexpr float SQRTHALF = 0.70710678118654752f;
}

// ------------------------- degree / dinv ----------------------------------
__global__ void k_fill_deg(float* deg) {
  int n = blockIdx.x * blockDim.x + threadIdx.x;
  if (n < NN) deg[n] = 1.0f;                       // self-loop
}
__global__ void k_deg(const int* __restrict__ dst, float* deg, int E) {
  int e = blockIdx.x * blockDim.x + threadIdx.x;
  if (e < E) atomicAdd(&deg[dst[e]], 1.0f);
}
__global__ void k_dinv(float* deg) {
  int n = blockIdx.x * blockDim.x + threadIdx.x;
  if (n < NN) deg[n] = rsqrtf(deg[n]);
}

// ------------------------- WMMA GEMM: Y = X @ W  (X:[2048,64] W:[64,64]) ---
// One wave computes one 16x16 tile; grid = (2048/16)*(64/16) = 512 waves.
__global__ void k_gemm_xw(const float* __restrict__ X, const float* __restrict__ W,
                          float* __restrict__ Y) {
  const int mt = blockIdx.x >> 2;         // 128 row tiles
  const int nt = blockIdx.x & 3;          // 4 col tiles
  const int lane = threadIdx.x;
  const int hl = lane >> 4;               // half-wave: 0 -> K pair {0,1}, 1 -> {2,3}
  const int ml = lane & 15;
  const int m0 = mt * 16, n0 = nt * 16;
#if HAS_WMMA_F32X4
  v8f c = {};
  for (int k = 0; k < D; k += 4) {
    const int kb = k + hl * 2;
    v2f a, b;
    a.x = X[(m0 + ml) * D + kb];          // A 16x4: lane holds row ml, 2 K values
    a.y = X[(m0 + ml) * D + kb + 1];
    b.x = W[kb * D + n0 + ml];            // B 4x16: lane holds col ml, 2 K values
    b.y = W[(kb + 1) * D + n0 + ml];
    c = __builtin_amdgcn_wmma_f32_16x16x4_f32(false, a, false, b, (short)0, c,
                                              false, false);
  }
  for (int r = 0; r < 8; ++r)             // D layout: vgpr r -> row r (+8 for hi half)
    Y[(m0 + r + hl * 8) * D + n0 + ml] = c[r];
#else
  for (int r = 0; r < 8; ++r) {
    const int row = m0 + r + hl * 8;
    float acc = 0.f;
    for (int k = 0; k < D; ++k) acc += X[row * D + k] * W[k * D + n0 + ml];
    Y[row * D + n0 + ml] = acc;
  }
#endif
}

// ---------------- WMMA GEMM: C[16,16] = A[16,K] @ B[K,16] + bias ----------
// Single wave; used for pooled@lp, pooled@cls, kf@kp (K = 64 or 1280).
__global__ void k_gemm16(const float* __restrict__ A, const float* __restrict__ B,
                         const float* __restrict__ bias, float* __restrict__ C,
                         int K) {
  const int lane = threadIdx.x;
  const int hl = lane >> 4;
  const int ml = lane & 15;
#if HAS_WMMA_F32X4
  v8f c = {};
  for (int k = 0; k < K; k += 4) {
    const int kb = k + hl * 2;
    v2f a, b;
    a.x = A[ml * K + kb];
    a.y = A[ml * K + kb + 1];
    b.x = B[kb * 16 + ml];
    b.y = B[(kb + 1) * 16 + ml];
    c = __builtin_amdgcn_wmma_f32_16x16x4_f32(false, a, false, b, (short)0, c,
                                              false, false);
  }
  for (int r = 0; r < 8; ++r)
    C[(r + hl * 8) * 16 + ml] = c[r] + bias[ml];
#else
  for (int r = 0; r < 8; ++r) {
    const int row = r + hl * 8;
    float acc = 0.f;
    for (int k = 0; k < K; ++k) acc += A[row * K + k] * B[k * 16 + ml];
    C[row * 16 + ml] = acc + bias[ml];
  }
#endif
}

// ------------------------- GCN aggregation --------------------------------
// out[n] = xw[n]*dinv[n]^2 + b   (self loop term + bias), then edge scatter.
__global__ void k_gcn_self(const float* __restrict__ xw, const float* __restrict__ dinv,
                           const float* __restrict__ b, float* __restrict__ out) {
  int t = blockIdx.x * blockDim.x + threadIdx.x;
  if (t < NN * D) {
    int n = t >> 6, dd = t & 63;
    float di = dinv[n];
    out[t] = xw[t] * di * di + b[dd];
  }
}
__global__ void k_gcn_edge(const int* __restrict__ src, const int* __restrict__ dst,
                           const float* __restrict__ xw, const float* __restrict__ dinv,
                           float* __restrict__ out, int E) {
  int t = blockIdx.x * blockDim.x + threadIdx.x;
  if (t < E * D) {
    int e = t >> 6, dd = t & 63;
    int s = src[e], r = dst[e];
    atomicAdd(&out[r * D + dd], xw[s * D + dd] * dinv[s] * dinv[r]);
  }
}

// ------------------------- global attention pool --------------------------
// One 128-thread block per graph; softmax over the 128 nodes.
__global__ void k_attpool(const float* __restrict__ X, const float* __restrict__ gW,
                          const float* __restrict__ gb, float* __restrict__ pooled) {
  __shared__ float lw[NPG];
  const int g = blockIdx.x;
  const int tid = threadIdx.x;            // 0..127, one node each
  const float* xg = X + (size_t)g * NPG * D;
  float logit = gb[0];
  for (int d = 0; d < D; ++d) logit += xg[tid * D + d] * gW[d];
  lw[tid] = logit;
  __syncthreads();
  float m = -3.4e38f;
  for (int j = 0; j < NPG; ++j) m = fmaxf(m, lw[j]);
  float e = __expf(logit - m);
  __syncthreads();
  lw[tid] = e;
  __syncthreads();
  float s = 0.f;
  for (int j = 0; j < NPG; ++j) s += lw[j];
  if (tid < D) {
    float acc = 0.f;
    for (int j = 0; j < NPG; ++j) acc += lw[j] * xg[j * D + tid];
    pooled[g * D + tid] = acc / s;
  }
}

// ------------------------- KDE soft-quantile features ---------------------
// One wave per (graph, dim): 16*64 = 1024 blocks of 32 threads.
__global__ void k_kde(const float* __restrict__ X, float* __restrict__ kf) {
  __shared__ float xs[NPG];
  __shared__ float dens[PP];
  __shared__ float bc[4];
  const int g = blockIdx.x >> 6;
  const int d = blockIdx.x & 63;
  const int lane = threadIdx.x;
  const float* xg = X + (size_t)g * NPG * D + d;

  // stage 128 values, wave-reduce min/max/sum/sumsq
  float mn = 3.4e38f, mx = -3.4e38f, sum = 0.f, sq = 0.f;
  for (int j = lane; j < NPG; j += 32) {
    float v = xg[j * D];
    xs[j] = v;
    mn = fminf(mn, v); mx = fmaxf(mx, v);
    sum += v; sq += v * v;
  }
  for (int off = 16; off > 0; off >>= 1) {
    mn = fminf(mn, __shfl_xor(mn, off, 32));
    mx = fmaxf(mx, __shfl_xor(mx, off, 32));
    sum += __shfl_xor(sum, off, 32);
    sq  += __shfl_xor(sq,  off, 32);
  }
  if (lane == 0) {
    float mean = sum * (1.f / NPG);
    float var  = fmaxf(sq * (1.f / NPG) - mean * mean, 0.f);
    float sd   = sqrtf(var) + (1e-8f / 3.f);
    bc[0] = mn - 1e-6f;
    bc[1] = mx + 1e-6f;
    bc[2] = 1.06f * sd * NPOW;            // Silverman bandwidth, n=128
  }
  __syncthreads();
  const float lo = bc[0], hi = bc[1], h = bc[2];
  const float step   = (hi - lo) * (1.f / (PP - 1));
  const float hinv   = 1.f / h;
  const float dscale = hinv * (1.f / (NPG * SQRT2PI));
  const float cs     = hinv * SQRTHALF;   // exp(-0.5*((g-x)/h)^2) = exp(-((g-x)*cs)^2)

  // pre-scale staged values (each lane rescales exactly the slots it wrote)
  for (int j = lane; j < NPG; j += 32) xs[j] *= cs;
  __syncthreads();

  // density: 4 grid points per staged-x broadcast read; 3 VALU + 1 TRANS per exp
  for (int p0 = lane; p0 < PP; p0 += 128) {   // uniform 4 iterations for all lanes
    const float g0 = (lo + step * (float)(p0 +  0)) * cs;
    const float g1 = (lo + step * (float)(p0 + 32)) * cs;
    const float g2 = (lo + step * (float)(p0 + 64)) * cs;
    const float g3 = (lo + step * (float)(p0 + 96)) * cs;
    float a0 = 0.f, a1 = 0.f, a2 = 0.f, a3 = 0.f;
    for (int j = 0; j < NPG; ++j) {
      const float xc = xs[j];
      float u0 = g0 - xc; a0 += __expf(-(u0 * u0));
      float u1 = g1 - xc; a1 += __expf(-(u1 * u1));
      float u2 = g2 - xc; a2 += __expf(-(u2 * u2));
      float u3 = g3 - xc; a3 += __expf(-(u3 * u3));
    }
    dens[p0] = a0 * dscale;
    if (p0 + 32 < PP) dens[p0 + 32] = a1 * dscale;
    if (p0 + 64 < PP) dens[p0 + 64] = a2 * dscale;
    if (p0 + 96 < PP) dens[p0 + 96] = a3 * dscale;
  }
  __syncthreads();

  // sequential cumsum (matches reference ordering), normalize by clipped total
  if (lane == 0) {
    float run = 0.f;
    for (int p = 0; p < PP; ++p) { run += dens[p]; dens[p] = run; }
    bc[3] = 1.f / fmaxf(run, 1e-8f);
  }
  __syncthreads();
  const float invT = bc[3];
  for (int p = lane; p < PP; p += 32) dens[p] *= invT;
  __syncthreads();

  // soft quantiles, register-blocked: single sweep over the grid, one LDS read
  // of cdf[p] feeds all 20 quantile sigmoids (40 accumulator VGPRs).
  float sw[QQ], sgw[QQ];
#pragma unroll
  for (int q = 0; q < QQ; ++q) { sw[q] = 0.f; sgw[q] = 0.f; }
  for (int p = lane; p < PP; p += 32) {
    const float cdf100 = dens[p] * 100.f;
    const float gp = lo + step * (float)p;
#pragma unroll
    for (int q = 0; q < QQ; ++q) {
      const float dist100 = fabsf(cdf100 - (float)q * (100.f / (QQ - 1)));
      const float w = 1.f / (1.f + __expf(dist100));   // sigmoid(-100*dist)
      sw[q]  += w;
      sgw[q] += gp * w;
    }
  }
  float* out = kf + (size_t)g * (D * QQ) + d * QQ;
#pragma unroll
  for (int q = 0; q < QQ; ++q) {
    float a = sw[q], b = sgw[q];
    for (int off = 16; off > 0; off >>= 1) {
      a += __shfl_xor(a, off, 32);
      b += __shfl_xor(b, off, 32);
    }
    if (lane == 0) out[q] = b / (a + 1e-8f);
  }
}

// ------------------------- final combine ----------------------------------
__global__ void k_combine(const float* __restrict__ hp0, const float* __restrict__ hp1,
                          const float* __restrict__ hp2, const float* __restrict__ kp0,
                          const float* __restrict__ kp1, const float* __restrict__ beta,
                          const float* __restrict__ h0, float* __restrict__ out) {
  int g = threadIdx.x;
  if (g < GG) {
    float acc = 0.f;
    for (int o = 0; o < DOUT; ++o) {
      float m = (hp0[g * DOUT + o] + hp1[g * DOUT + o] + hp2[g * DOUT + o]) * (1.f / 3.f);
      float k = (kp0[g * DOUT + o] + kp1[g * DOUT + o]) * 0.5f;
      acc += (m + k) * beta[o];
    }
    out[g] = acc + h0[0];
  }
}

// ---------------------------------------------------------------------------
extern "C" void kernel_launch(void* const* d_in, const int* in_sizes, int n_in,
                              void* d_out, int out_size, void* d_ws, size_t ws_size,
                              hipStream_t stream) {
  const float* x       = (const float*)d_in[0];
  const int*   ei      = (const int*)  d_in[1];
  const float* gcn_W0  = (const float*)d_in[2];
  const float* gcn_b0  = (const float*)d_in[3];
  const float* gcn_W1  = (const float*)d_in[4];
  const float* gcn_b1  = (const float*)d_in[5];
  const float* lp_W0   = (const float*)d_in[6];
  const float* lp_b0   = (const float*)d_in[7];
  const float* lp_W1   = (const float*)d_in[8];
  const float* lp_b1   = (const float*)d_in[9];
  const float* cls_W   = (const float*)d_in[10];
  const float* cls_b   = (const float*)d_in[11];
  const float* kp_W0   = (const float*)d_in[12];
  const float* kp_b0   = (const float*)d_in[13];
  const float* kp_W1   = (const float*)d_in[14];
  const float* kp_b1   = (const float*)d_in[15];
  const float* gate_W0 = (const float*)d_in[16];
  const float* gate_b0 = (const float*)d_in[17];
  const float* gate_W1 = (const float*)d_in[18];
  const float* gate_b1 = (const float*)d_in[19];
  const float* gate_W2 = (const float*)d_in[20];
  const float* gate_b2 = (const float*)d_in[21];
  const float* beta    = (const float*)d_in[22];
  const float* h0s     = (const float*)d_in[23];
  float* out = (float*)d_out;

  const int E = in_sizes[1] / 2;
  const int* src = ei;
  const int* dst = ei + E;

  // workspace carve-up (floats)
  float* w = (float*)d_ws;
  float* dinv   = w;              w += NN;           // 2048
  float* xw     = w;              w += NN * D;       // 131072
  float* cur1   = w;              w += NN * D;
  float* cur2   = w;              w += NN * D;
  float* pooled0= w;              w += GG * D;
  float* pooled1= w;              w += GG * D;
  float* pooled2= w;              w += GG * D;
  float* hp0    = w;              w += GG * DOUT;
  float* hp1    = w;              w += GG * DOUT;
  float* hp2    = w;              w += GG * DOUT;
  float* kf0    = w;              w += GG * D * QQ;  // 20480
  float* kf1    = w;              w += GG * D * QQ;
  float* kp0    = w;              w += GG * DOUT;
  float* kp1    = w;              w += GG * DOUT;

  // --- degrees (shared by both GCN layers)
  k_fill_deg<<<(NN + 255) / 256, 256, 0, stream>>>(dinv);
  k_deg<<<(E + 255) / 256, 256, 0, stream>>>(dst, dinv, E);
  k_dinv<<<(NN + 255) / 256, 256, 0, stream>>>(dinv);

  // --- layer 0 (cur = x)
  k_attpool<<<GG, NPG, 0, stream>>>(x, gate_W0, gate_b0, pooled0);
  k_gemm16<<<1, 32, 0, stream>>>(pooled0, lp_W0, lp_b0, hp0, D);
  k_kde<<<GG * D, 32, 0, stream>>>(x, kf0);
  k_gemm16<<<1, 32, 0, stream>>>(kf0, kp_W0, kp_b0, kp0, D * QQ);
  k_gemm_xw<<<(NN / 16) * (D / 16), 32, 0, stream>>>(x, gcn_W0, xw);
  k_gcn_self<<<(NN * D + 255) / 256, 256, 0, stream>>>(xw, dinv, gcn_b0, cur1);
  k_gcn_edge<<<(E * D + 255) / 256, 256, 0, stream>>>(src, dst, xw, dinv, cur1, E);

  // --- layer 1 (cur = cur1)
  k_attpool<<<GG, NPG, 0, stream>>>(cur1, gate_W1, gate_b1, pooled1);
  k_gemm16<<<1, 32, 0, stream>>>(pooled1, lp_W1, lp_b1, hp1, D);
  k_kde<<<GG * D, 32, 0, stream>>>(cur1, kf1);
  k_gemm16<<<1, 32, 0, stream>>>(kf1, kp_W1, kp_b1, kp1, D * QQ);
  k_gemm_xw<<<(NN / 16) * (D / 16), 32, 0, stream>>>(cur1, gcn_W1, xw);
  k_gcn_self<<<(NN * D + 255) / 256, 256, 0, stream>>>(xw, dinv, gcn_b1, cur2);
  k_gcn_edge<<<(E * D + 255) / 256, 256, 0, stream>>>(src, dst, xw, dinv, cur2, E);

  // --- classifier head on final features
  k_attpool<<<GG, NPG, 0, stream>>>(cur2, gate_W2, gate_b2, pooled2);
  k_gemm16<<<1, 32, 0, stream>>>(pooled2, cls_W, cls_b, hp2, D);

  // --- combine
  k_combine<<<1, 32, 0, stream>>>(hp0, hp1, hp2, kp0, kp1, beta, h0s, out);

  (void)n_in; (void)out_size; (void)ws_size;
}